// MSDeformAttn_10136122818970
// MI455X (gfx1250) — compile-verified
//
#include <hip/hip_runtime.h>
#include <hip/hip_bf16.h>
#include <math.h>

typedef float v2f __attribute__((ext_vector_type(2)));
typedef float v8f __attribute__((ext_vector_type(8)));

#define NHEADS  8
#define NLEVELS 4
#define NPOINTS 4
#define DH      32
#define CMODEL  256
#define KDIM    256     // all GEMMs here have K = 256
#define LQ      21760   // 128*128 + 64*64 + 32*32 + 16*16

// ---------------------------------------------------------------------------
// GEMM: out[M,Nc] = A[M,K=256] @ W[256,Nc] + bias[Nc], fp32 WMMA 16x16x4.
// Block = 256 threads (8 waves). Block tile: 16 rows x 128 cols; wave w owns
// the 16x16 tile at columns blockIdx.x*128 + w*16.
// A tile (16x256 = 16KB) is staged in LDS once per block with coalesced
// 8B/lane loads, stored as float2 pairs [kp][m] with row stride 17 so both
// the staging stores and the per-k fragment reads are LDS-bank conflict-free.
// Fragment layouts (per ISA):
//   A-frag (16x4 f32): lane L -> M=L&15, VGPR v -> K = v + 2*(L>>4)
//   B-frag (4x16 f32): lane L -> N=L&15, VGPR v -> K = v + 2*(L>>4)
//   C/D   (16x16 f32): lane L -> col L&15, VGPR v -> row v + 8*(L>>4)
// ---------------------------------------------------------------------------
__global__ __launch_bounds__(256) void gemm_bias_wmma(
    const float* __restrict__ A, const float* __restrict__ W,
    const float* __restrict__ bias, float* __restrict__ out,
    int M, int Nc)
{
  __shared__ v2f aS[(KDIM / 2) * 17 + 16];   // [kp]*17 + m, kp in [0,128)

  const int lane  = threadIdx.x & 31;
  const int wave  = threadIdx.x >> 5;
  const int nbase = blockIdx.x * 128 + wave * 16;
  const int mbase = blockIdx.y * 16;

  // ---- stage A tile: 16 rows x 128 float2, coalesced along k ----
  for (int i = threadIdx.x; i < 16 * (KDIM / 2); i += 256) {
    const int m  = i >> 7;        // 0..15
    const int kp = i & 127;       // float2 pair index along K
    v2f d = *(const v2f*)&A[(size_t)(mbase + m) * KDIM + 2 * kp];
    aS[kp * 17 + m] = d;
  }
  // prefetch the next block's A tile while we compute (global_prefetch_b8)
  if (mbase + 16 < M) {
    __builtin_prefetch(&A[(size_t)(mbase + 16) * KDIM + (threadIdx.x << 3)], 0, 1);
  }
  __syncthreads();

  const int hi   = lane >> 4;          // 0: lanes 0-15, 1: lanes 16-31
  const int mloc = lane & 15;
  const int ncol = nbase + (lane & 15);

  v8f acc = {};
  for (int k = 0; k < KDIM; k += 4) {
    const int ka = k + 2 * hi;                 // even
    v2f a = aS[(ka >> 1) * 17 + mloc];         // {A[m][ka], A[m][ka+1]}
    v2f b;
    b.x = W[(size_t)ka * Nc + ncol];
    b.y = W[(size_t)(ka + 1) * Nc + ncol];
    acc = __builtin_amdgcn_wmma_f32_16x16x4_f32(
        /*neg_a=*/false, a, /*neg_b=*/false, b,
        /*c_mod=*/(short)0, acc, /*reuse_a=*/false, /*reuse_b=*/false);
  }

  const float bcol = bias[ncol];
#pragma unroll
  for (int v = 0; v < 8; ++v) {
    const int row = mbase + v + 8 * hi;
    out[(size_t)row * Nc + ncol] = acc[v] + bcol;
  }
}

// ---------------------------------------------------------------------------
// In-place softmax over contiguous groups of 16 (attention weights).
// Group t corresponds to (n,q,h): buffer layout (n*Lq+q)*128 + h*16 + lp.
// ---------------------------------------------------------------------------
__global__ __launch_bounds__(256) void softmax16_kernel(float* __restrict__ buf, int total)
{
  const int t = blockIdx.x * blockDim.x + threadIdx.x;
  if (t >= total) return;
  float* p = buf + (size_t)t * 16;
  float v[16];
  float m = -INFINITY;
#pragma unroll
  for (int i = 0; i < 16; ++i) { v[i] = p[i]; m = fmaxf(m, v[i]); }
  float s = 0.f;
#pragma unroll
  for (int i = 0; i < 16; ++i) { v[i] = __expf(v[i] - m); s += v[i]; }
  const float inv = 1.f / s;
#pragma unroll
  for (int i = 0; i < 16; ++i) p[i] = v[i] * inv;
}

// ---------------------------------------------------------------------------
// In-place: loc = reference_points + off / normalizer
// buffer layout (N,Lq,H,L,P,2); element c = h*32 + l*8 + p*2 + xy
// normalizer: xy==0 -> W_l (shapes[2l+1]), xy==1 -> H_l (shapes[2l])
// ---------------------------------------------------------------------------
__global__ __launch_bounds__(256) void loc_kernel(
    float* __restrict__ loc, const float* __restrict__ ref,
    const int* __restrict__ shapes, size_t total)
{
  const size_t i = (size_t)blockIdx.x * blockDim.x + threadIdx.x;
  if (i >= total) return;
  const int    c  = (int)(i & 255);
  const size_t nq = i >> 8;
  const int l  = (c >> 3) & 3;
  const int xy = c & 1;
  const float norm = (float)((xy == 0) ? shapes[2 * l + 1] : shapes[2 * l]);
  const float r = ref[(nq * NLEVELS + l) * 2 + xy];
  loc[i] = r + loc[i] / norm;
}

// ---------------------------------------------------------------------------
// Bilinear sampling + attention-weighted reduction.
// One wave per (n,q,h); lane = channel d in [0,32).
// value: (N, Lv, H, Dh)  -> head channels contiguous => 128B coalesced gathers
// samp : (N, Lq, H*Dh)   -> feeds output GEMM directly
// The whole value tensor (44.6 MB) is L2-resident (192 MB L2), so the 2.9 GB
// of corner gathers stream from L2, not HBM.
// ---------------------------------------------------------------------------
__device__ __forceinline__ float corner_tap(const float* __restrict__ vbase,
                                            int yi, int xi, int Hl, int Wl, float wgt)
{
  const float valid = (yi >= 0 && yi < Hl && xi >= 0 && xi < Wl) ? 1.f : 0.f;
  int cy = yi < 0 ? 0 : (yi > Hl - 1 ? Hl - 1 : yi);
  int cx = xi < 0 ? 0 : (xi > Wl - 1 ? Wl - 1 : xi);
  const float g = vbase[(size_t)(cy * Wl + cx) * (NHEADS * DH)];
  return g * (wgt * valid);
}

__global__ __launch_bounds__(256) void sample_kernel(
    const float* __restrict__ value, const float* __restrict__ loc,
    const float* __restrict__ aw, const int* __restrict__ shapes,
    const int* __restrict__ starts, float* __restrict__ samp, int total_nqh)
{
  const int w = blockIdx.x * 8 + (threadIdx.x >> 5);
  if (w >= total_nqh) return;
  const int lane = threadIdx.x & 31;          // channel d
  const int h  = w & 7;
  const int nq = w >> 3;                      // n*Lq + q
  const int n  = nq / LQ;

  const float* locp = loc + (size_t)(nq * NHEADS + h) * (NLEVELS * NPOINTS * 2);
  const float* awp  = aw  + (size_t)(nq * NHEADS + h) * (NLEVELS * NPOINTS);

  float acc = 0.f;
#pragma unroll
  for (int l = 0; l < NLEVELS; ++l) {
    const int Hl = shapes[2 * l], Wl = shapes[2 * l + 1];
    const int s0 = starts[l];
    const float* vbase =
        value + (((size_t)n * LQ + s0) * NHEADS + h) * DH + lane;
#pragma unroll
    for (int p = 0; p < NPOINTS; ++p) {
      const float lx = locp[(l * NPOINTS + p) * 2 + 0];
      const float ly = locp[(l * NPOINTS + p) * 2 + 1];
      const float a  = awp[l * NPOINTS + p];
      const float x = lx * (float)Wl - 0.5f;
      const float y = ly * (float)Hl - 0.5f;
      const float x0f = floorf(x), y0f = floorf(y);
      const float fx = x - x0f, fy = y - y0f;
      const int x0 = (int)x0f, y0 = (int)y0f;
      float s;
      s  = corner_tap(vbase, y0,     x0,     Hl, Wl, (1.f - fy) * (1.f - fx));
      s += corner_tap(vbase, y0,     x0 + 1, Hl, Wl, (1.f - fy) * fx);
      s += corner_tap(vbase, y0 + 1, x0,     Hl, Wl, fy * (1.f - fx));
      s += corner_tap(vbase, y0 + 1, x0 + 1, Hl, Wl, fy * fx);
      acc += a * s;
    }
  }
  samp[(size_t)nq * CMODEL + h * DH + lane] = acc;
}

// ---------------------------------------------------------------------------
extern "C" void kernel_launch(void* const* d_in, const int* in_sizes, int n_in,
                              void* d_out, int out_size, void* d_ws, size_t ws_size,
                              hipStream_t stream)
{
  const float* query  = (const float*)d_in[0];
  const float* refpt  = (const float*)d_in[1];
  const float* inflat = (const float*)d_in[2];
  const int*   shapes = (const int*)d_in[3];
  const int*   starts = (const int*)d_in[4];
  const float* W_off  = (const float*)d_in[5];
  const float* b_off  = (const float*)d_in[6];
  const float* W_attn = (const float*)d_in[7];
  const float* b_attn = (const float*)d_in[8];
  const float* W_val  = (const float*)d_in[9];
  const float* b_val  = (const float*)d_in[10];
  const float* W_out  = (const float*)d_in[11];
  const float* b_out  = (const float*)d_in[12];

  const int C = CMODEL;
  const int M = in_sizes[0] / C;            // N * Lq = 43520

  float* out = (float*)d_out;               // (N,Lq,C)
  float* loc = out + (size_t)M * C;         // (N,Lq,H,L,P,2) second tuple output

  float* valuebuf = (float*)d_ws;                         // M*256 f32
  float* sampbuf  = valuebuf + (size_t)M * C;             // M*256 f32
  float* awbuf    = sampbuf  + (size_t)M * C;             // M*128 f32

  const dim3 gblk(256);
  const int mtiles = M / 16;

  // 1) value projection: input_flatten @ W_val + b_val
  gemm_bias_wmma<<<dim3(C / 128, mtiles), gblk, 0, stream>>>(
      inflat, W_val, b_val, valuebuf, M, C);

  // 2) attention logits: query @ W_attn + b_attn (128 cols)
  gemm_bias_wmma<<<dim3(1, mtiles), gblk, 0, stream>>>(
      query, W_attn, b_attn, awbuf, M, 128);

  // 3) offsets straight into the loc output slot
  gemm_bias_wmma<<<dim3(C / 128, mtiles), gblk, 0, stream>>>(
      query, W_off, b_off, loc, M, C);

  // 4) softmax over 16 (level,point) logits per (n,q,h), in place
  const int totalNH = M * NHEADS;
  softmax16_kernel<<<(totalNH + 255) / 256, 256, 0, stream>>>(awbuf, totalNH);

  // 5) loc = ref + off / normalizer, in place in d_out's second half
  const size_t totalLoc = (size_t)M * C;
  loc_kernel<<<(unsigned)((totalLoc + 255) / 256), 256, 0, stream>>>(
      loc, refpt, shapes, totalLoc);

  // 6) bilinear sampling + attention reduction (L2-resident gathers)
  sample_kernel<<<(totalNH + 7) / 8, 256, 0, stream>>>(
      valuebuf, loc, awbuf, shapes, starts, sampbuf, totalNH);

  // 7) output projection into d_out's first half
  gemm_bias_wmma<<<dim3(C / 128, mtiles), gblk, 0, stream>>>(
      sampbuf, W_out, b_out, out, M, C);
}